// gnn_v4_53652731461900
// MI455X (gfx1250) — compile-verified
//
#include <hip/hip_runtime.h>

typedef __attribute__((ext_vector_type(2))) float v2f;
typedef __attribute__((ext_vector_type(8))) float v8f;

#define NNODES 100000
#define NEDGES 3200000
#define FDIM   128
#define CDIM   32

// ---------------------------------------------------------------------------
// GEMM: out[n x 32] = x[n x K] @ W[K x 32]     (fp32 via V_WMMA_F32_16X16X4_F32)
// EPILOGUE=true additionally computes relu(acc + agg + bias).
// One wave computes a 16x32 tile (two 16x16 f32 accumulators).
// ---------------------------------------------------------------------------
template <bool EPILOGUE>
__global__ __launch_bounds__(256) void gemm_wmma_kernel(
    const float* __restrict__ x, const float* __restrict__ W,
    const float* __restrict__ agg, const float* __restrict__ bias,
    float* __restrict__ out, int n, int K) {
  const int lane = threadIdx.x & 31;
  const int wave = threadIdx.x >> 5;
  const int tile = blockIdx.x * 8 + wave;   // 16-row output tile per wave
  const int numTiles = n >> 4;              // n == 100000 -> exactly 6250 tiles
  if (tile >= numTiles) return;             // wave-uniform exit; EXEC stays all-1s

  const int row0 = tile << 4;
  const int mrow = row0 + (lane & 15);      // A row for this lane
  const int kadd = (lane < 16) ? 0 : 2;     // ISA 16x4 f32 A layout: lanes 16..31 hold K+2,K+3
  const int col  = lane & 15;

  v8f acc0 = {};
  v8f acc1 = {};

  const int ksteps = K >> 2;
  for (int ks = 0; ks < ksteps; ++ks) {
    const int kb = (ks << 2) + kadd;

    // A fragment: two consecutive K values for this lane's row
    const float* ap = x + (size_t)mrow * K + kb;
    v2f a;
    a.x = ap[0];
    a.y = ap[1];

    // B fragments (rows kb, kb+1 of W) for column tiles 0 and 16
    const float* bp = W + (size_t)kb * CDIM + col;
    v2f b0, b1;
    b0.x = bp[0];
    b0.y = bp[CDIM];
    b1.x = bp[16];
    b1.y = bp[CDIM + 16];

    acc0 = __builtin_amdgcn_wmma_f32_16x16x4_f32(false, a, false, b0,
                                                 (short)0, acc0, false, false);
    acc1 = __builtin_amdgcn_wmma_f32_16x16x4_f32(false, a, false, b1,
                                                 (short)0, acc1, false, false);
  }

  // C/D layout: VGPR j -> row = row0 + j (lanes 0..15) / row0 + j + 8 (lanes 16..31)
  const int rbase = row0 + ((lane < 16) ? 0 : 8);
#pragma unroll
  for (int j = 0; j < 8; ++j) {
    const int r = rbase + j;
    float v0 = acc0[j];
    float v1 = acc1[j];
    if (EPILOGUE) {
      v0 += agg[(size_t)r * CDIM + col] + bias[col];
      v1 += agg[(size_t)r * CDIM + col + 16] + bias[col + 16];
      v0 = fmaxf(v0, 0.0f);
      v1 = fmaxf(v1, 0.0f);
    }
    out[(size_t)r * CDIM + col] = v0;
    out[(size_t)r * CDIM + col + 16] = v1;
  }
}

// ---------------------------------------------------------------------------
// Edge scatter: agg[dst[e], c] += h[src[e], c] * w[e]   (one wave per edge)
// Atomics land in the 12.8 MB agg table -> resolve in L2 (192 MB), not HBM.
// ---------------------------------------------------------------------------
__global__ __launch_bounds__(256) void scatter_kernel(
    const float* __restrict__ h, const int* __restrict__ src,
    const int* __restrict__ dst, const float* __restrict__ w,
    float* __restrict__ agg, int E) {
  const int e = blockIdx.x * 8 + (threadIdx.x >> 5);
  if (e >= E) return;
  const int lane = threadIdx.x & 31;
  const int s = src[e];
  const int d = dst[e];
  const float wv = w[e];
  const float val = h[(size_t)s * CDIM + lane] * wv;
  atomicAdd(&agg[(size_t)d * CDIM + lane], val);
}

__global__ __launch_bounds__(256) void zero_kernel(float* __restrict__ p, int n) {
  const int i = blockIdx.x * blockDim.x + threadIdx.x;
  if (i < n) p[i] = 0.0f;
}

// Global sum pool over rows: pooled[c] = sum_r x[r, c]
__global__ __launch_bounds__(256) void pool_kernel(
    const float* __restrict__ xin, float* __restrict__ pooled, int n) {
  __shared__ float red[256];
  const int c = threadIdx.x & 31;
  const int w = threadIdx.x >> 5;
  float s = 0.0f;
  for (int r = blockIdx.x * 8 + w; r < n; r += gridDim.x * 8)
    s += xin[(size_t)r * CDIM + c];
  red[threadIdx.x] = s;
  __syncthreads();
  if (threadIdx.x < 32) {
    float t = 0.0f;
#pragma unroll
    for (int i = 0; i < 8; ++i) t += red[i * 32 + c];
    atomicAdd(&pooled[c], t);
  }
}

// out = (pooled @ Wd1 + bd1) @ Wd2 + bd2   (32 -> 24 -> 1; trivial)
__global__ void dense_kernel(const float* __restrict__ pooled,
                             const float* __restrict__ Wd1,
                             const float* __restrict__ bd1,
                             const float* __restrict__ Wd2,
                             const float* __restrict__ bd2,
                             float* __restrict__ out) {
  if (threadIdx.x == 0 && blockIdx.x == 0) {
    float acc = bd2[0];
    for (int h = 0; h < 24; ++h) {
      float t = bd1[h];
      for (int c = 0; c < 32; ++c) t += pooled[c] * Wd1[c * 24 + h];
      acc += t * Wd2[h];
    }
    out[0] = acc;
  }
}

extern "C" void kernel_launch(void* const* d_in, const int* in_sizes, int n_in,
                              void* d_out, int out_size, void* d_ws, size_t ws_size,
                              hipStream_t stream) {
  (void)in_sizes; (void)n_in; (void)out_size; (void)ws_size;
  const float* x    = (const float*)d_in[0];
  const int*   esrc = (const int*)d_in[1];
  const int*   edst = (const int*)d_in[2];
  const float* ew   = (const float*)d_in[3];
  const float* K1a  = (const float*)d_in[4];
  const float* K2a  = (const float*)d_in[5];
  const float* ba   = (const float*)d_in[6];
  const float* K1b  = (const float*)d_in[7];
  const float* K2b  = (const float*)d_in[8];
  const float* bb   = (const float*)d_in[9];
  const float* Wd1  = (const float*)d_in[10];
  const float* bd1  = (const float*)d_in[11];
  const float* Wd2  = (const float*)d_in[12];
  const float* bd2  = (const float*)d_in[13];
  float* out = (float*)d_out;

  float* ws = (float*)d_ws;
  const int NC = NNODES * CDIM;
  float* b0     = ws;            // h (layer a), then h2, then final x2
  float* b1     = ws + (size_t)NC;      // agg (both layers)
  float* b2     = ws + (size_t)2 * NC;  // x1 (layer-a output)
  float* pooled = ws + (size_t)3 * NC;  // 32 floats

  const dim3 blk(256);
  const int gemmGrid = (NNODES / 16 + 7) / 8;  // 6250 tiles / 8 waves per block
  const int edgeGrid = (NEDGES + 7) / 8;       // one wave per edge
  const int zeroGrid = (NC + 255) / 256;

  // ---- Layer a: x1 = relu(A_norm @ (x @ K1a) + x @ K2a + ba)
  gemm_wmma_kernel<false><<<gemmGrid, blk, 0, stream>>>(x, K1a, nullptr, nullptr, b0, NNODES, FDIM);
  zero_kernel<<<zeroGrid, blk, 0, stream>>>(b1, NC);
  scatter_kernel<<<edgeGrid, blk, 0, stream>>>(b0, esrc, edst, ew, b1, NEDGES);
  gemm_wmma_kernel<true><<<gemmGrid, blk, 0, stream>>>(x, K2a, b1, ba, b2, NNODES, FDIM);

  // ---- Layer b: x2 = relu(A_norm @ (x1 @ K1b) + x1 @ K2b + bb)
  gemm_wmma_kernel<false><<<gemmGrid, blk, 0, stream>>>(b2, K1b, nullptr, nullptr, b0, NNODES, CDIM);
  zero_kernel<<<zeroGrid, blk, 0, stream>>>(b1, NC);
  scatter_kernel<<<edgeGrid, blk, 0, stream>>>(b0, esrc, edst, ew, b1, NEDGES);
  gemm_wmma_kernel<true><<<gemmGrid, blk, 0, stream>>>(b2, K2b, b1, bb, b0, NNODES, CDIM);

  // ---- Pool + dense head
  zero_kernel<<<1, 32, 0, stream>>>(pooled, 32);
  pool_kernel<<<512, blk, 0, stream>>>(b0, pooled, NNODES);
  dense_kernel<<<1, 32, 0, stream>>>(pooled, Wd1, bd1, Wd2, bd2, out);
}